// YOLOLoss_44994077393252
// MI455X (gfx1250) — compile-verified
//
#include <hip/hip_runtime.h>
#include <stdint.h>

// ---------------- problem constants (from reference) ----------------
#define IN_HH 76
#define IN_WW 76
#define HW    (IN_HH * IN_WW)          // 5776
#define BS    32
#define NA    3                        // anchors used at this level (SUB..SUB+2)
#define NT    40                       // targets per batch
#define CPB   (NA * HW)                // 17328 cells per batch
#define BLK   256
#define BLOCKS_PER_BATCH ((CPB + BLK - 1) / BLK)   // 68
#define NBLK  (BS * BLOCKS_PER_BATCH)              // 2176
#define SUBA  6
#define BCE_EPS 1e-7f
#define GTS   6                        // floats per gt record in LDS

// anchors / STRIDE, STRIDE = 608/76 = 8
__device__ __constant__ float c_aw[9] = {10.f/8.f, 16.f/8.f, 33.f/8.f, 30.f/8.f, 62.f/8.f,
                                         59.f/8.f, 116.f/8.f, 156.f/8.f, 373.f/8.f};
__device__ __constant__ float c_ah[9] = {13.f/8.f, 30.f/8.f, 23.f/8.f, 61.f/8.f, 45.f/8.f,
                                         119.f/8.f, 90.f/8.f, 198.f/8.f, 326.f/8.f};

__device__ __forceinline__ float sigmoidf(float v) { return 1.0f / (1.0f + expf(-v)); }

__device__ __forceinline__ float bce(float p, float t) {
    p = fminf(fmaxf(p, BCE_EPS), 1.0f - BCE_EPS);
    return -t * logf(p) - (1.0f - t) * logf(1.0f - p);
}

// ---------------- K1: per-target preprocessing ----------------
// gt : [BS*NT][6]  = {gx1, gy1, gx2, gy2, area, cellid-bits}  (ALL targets; cellid=-1 if invalid)
// rec: [BS*NT][12] = {tx, ty, tw, th, bls, tcls0, yaw, pitch, roll, pad...}
__global__ void k_prep(const float* __restrict__ tgt, float* __restrict__ gt,
                       float* __restrict__ rec) {
    int idx = blockIdx.x * blockDim.x + threadIdx.x;
    if (idx >= BS * NT) return;
    const float* tp = tgt + (size_t)idx * 8;
    float t0 = tp[0], t1 = tp[1], t2 = tp[2], t3 = tp[3];
    float t4 = tp[4], t5 = tp[5], t6 = tp[6], t7 = tp[7];
    float gx = t0 * IN_WW, gy = t1 * IN_HH, gw = t2 * IN_WW, gh = t3 * IN_HH;

    // argmax anchor IoU (first-max wins, matching jnp.argmax)
    int best = 0; float bestv = -1.0f;
    #pragma unroll
    for (int k = 0; k < 9; k++) {
        float aw = c_aw[k], ah = c_ah[k];
        float inter = fminf(gw, aw) * fminf(gh, ah);
        float uni   = gw * gh + aw * ah - inter;
        float r     = inter / uni;
        if (r > bestv) { bestv = r; best = k; }
    }

    int gi = (int)floorf(gx), gj = (int)floorf(gy);
    bool valid = (best >= SUBA) && (best < SUBA + NA) &&
                 gi >= 0 && gi < IN_WW && gj >= 0 && gj < IN_HH;
    int cellid = valid ? (((best - SUBA) * IN_HH + gj) * IN_WW + gi) : -1;

    float* g = gt + (size_t)idx * GTS;
    g[0] = gx - 0.5f * gw; g[1] = gy - 0.5f * gh;
    g[2] = gx + 0.5f * gw; g[3] = gy + 0.5f * gh;
    g[4] = gw * gh;
    g[5] = __int_as_float(cellid);

    float* r9 = rec + (size_t)idx * 12;
    r9[0] = gx - (float)gi;
    r9[1] = gy - (float)gj;
    r9[2] = logf(gw / c_aw[best]);
    r9[3] = logf(gh / c_ah[best]);
    r9[4] = 2.0f - t2 * t3;                    // bls = 2 - bls_x*bls_y
    r9[5] = ((int)t4 == 0) ? 1.0f : 0.0f;      // tcls (NUM_CLASSES==1)
    r9[6] = t5; r9[7] = t6; r9[8] = t7;
}

// ---------------- K2: per-cell loss, one block handles 256 cells of one batch ----------------
__global__ void __launch_bounds__(BLK) k_main(const float* __restrict__ inp,
                                              const float* __restrict__ gt,
                                              const float* __restrict__ rec,
                                              float* __restrict__ part) {
    __shared__ __align__(16) float sGT[NT * GTS];   // 240 floats = 960 B
    __shared__ float sRed[8][10];

    int tid = threadIdx.x;
    int b   = blockIdx.x / BLOCKS_PER_BATCH;
    int cib = (blockIdx.x % BLOCKS_PER_BATCH) * BLK + tid;

    // CDNA5 async-tensor path: stage this batch's 40 GT records into LDS.
    // 240 floats = 60 x b128 transfers, lanes tid<60 each move 16B.
    const float* gsrc = gt + (size_t)b * (NT * GTS);
    if (tid < (NT * GTS * 4) / 16) {
        uint32_t ldsa = (uint32_t)(uintptr_t)(&sGT[0]) + (uint32_t)tid * 16u;
        uint32_t voff = (uint32_t)tid * 16u;
        asm volatile("global_load_async_to_lds_b128 %0, %1, %2"
                     :: "v"(ldsa), "v"(voff), "s"(gsrc) : "memory");
    }

    float acc[10];
    #pragma unroll
    for (int k = 0; k < 10; k++) acc[k] = 0.0f;

    bool active = (cib < CPB);
    int a = 0, j = 0, i = 0;
    const float* base = inp;
    if (active) {
        a = cib / HW;
        int r = cib % HW;
        j = r / IN_WW; i = r % IN_WW;
        base = inp + ((size_t)b * 27 + (size_t)a * 9) * HW + (size_t)j * IN_WW + i;
        // cover the 9 strided channel-plane loads while the async copy is in flight
        __builtin_prefetch(base, 0, 3);
        __builtin_prefetch(base + 4 * HW, 0, 3);
        __builtin_prefetch(base + 8 * HW, 0, 3);
    }

    asm volatile("s_wait_asynccnt 0" ::: "memory");
    __syncthreads();

    if (active) {
        float c0 = base[0],      c1 = base[HW],     c2 = base[2 * HW];
        float c3 = base[3 * HW], c4 = base[4 * HW], c5 = base[5 * HW];
        float c6 = base[6 * HW], c7 = base[7 * HW], c8 = base[8 * HW];

        float x    = sigmoidf(c0), y = sigmoidf(c1);
        float conf = sigmoidf(c4), cls = sigmoidf(c5);
        float yaw  = tanhf(c6), pit = tanhf(c7), rol = tanhf(c8);

        // predicted box (uses level anchors SUBA..SUBA+2)
        float px = x + (float)i, py = y + (float)j;
        float pw = expf(c2) * c_aw[SUBA + a];
        float ph = expf(c3) * c_ah[SUBA + a];
        float px1 = px - 0.5f * pw, py1 = py - 0.5f * ph;
        float px2 = px + 0.5f * pw, py2 = py + 0.5f * ph;
        float areap = pw * ph;

        // fused IoU-max + ownership scan (ascending t => JAX last-write-wins scatter)
        float ioumax = 0.0f;
        int   own    = -1;
        #pragma unroll 8
        for (int t = 0; t < NT; t++) {
            float gx1 = sGT[t * GTS + 0], gy1 = sGT[t * GTS + 1];
            float gx2 = sGT[t * GTS + 2], gy2 = sGT[t * GTS + 3];
            float ag  = sGT[t * GTS + 4];
            int   tc  = __float_as_int(sGT[t * GTS + 5]);
            float iw = fmaxf(fminf(px2, gx2) - fmaxf(px1, gx1), 0.0f);
            float ih = fmaxf(fminf(py2, gy2) - fmaxf(py1, gy1), 0.0f);
            float inter = iw * ih;
            float iou = inter / (areap + ag - inter);
            ioumax = (iou > ioumax) ? iou : ioumax;
            if (tc == cib) own = t;
        }

        if (own >= 0) {
            const float* rr = rec + ((size_t)b * NT + own) * 12;
            float tx = rr[0], ty = rr[1], tw = rr[2], th = rr[3];
            float bls = rr[4], tc = rr[5], tyaw = rr[6], tpit = rr[7], trol = rr[8];
            acc[0] = bce(x, tx) * bls;
            acc[1] = bce(y, ty) * bls;
            float dw = c2 - tw, dh = c3 - th;
            acc[2] = 0.5f * dw * dw * bls;
            acc[3] = 0.5f * dh * dh * bls;
            acc[4] = bce(conf, 1.0f);          // mask term; noobj==0 at owner cells
            acc[5] = bce(cls, tc);
            float d = yaw - tyaw; float ad = 2.0f - fabsf(d);
            acc[6] = fminf(d * d, ad * ad);
            float dp = pit - tpit, dr = rol - trol;
            acc[7] = dp * dp;
            acc[8] = dr * dr;
            acc[9] = 1.0f;                     // mask (distinct-cell) count
        } else if (!(ioumax > 0.5f)) {
            acc[4] = bce(conf, 0.0f);          // noobj term
        }
    }

    // wave32 shuffle reduce, then cross-wave via LDS
    #pragma unroll
    for (int k = 0; k < 10; k++) {
        float v = acc[k];
        #pragma unroll
        for (int off = 16; off > 0; off >>= 1) v += __shfl_down(v, off, 32);
        acc[k] = v;
    }
    int lane = tid & 31, wv = tid >> 5;
    if (lane == 0) {
        #pragma unroll
        for (int k = 0; k < 10; k++) sRed[wv][k] = acc[k];
    }
    __syncthreads();
    if (tid < 10) {
        float s = 0.0f;
        #pragma unroll
        for (int w = 0; w < 8; w++) s += sRed[w][tid];
        part[(size_t)blockIdx.x * 10 + tid] = s;
    }
}

// ---------------- K3: deterministic final reduction -> 12 outputs ----------------
__global__ void __launch_bounds__(BLK) k_final(const float* __restrict__ part,
                                               float* __restrict__ out) {
    __shared__ float sAcc[10];
    __shared__ float sW[8];
    int tid = threadIdx.x;
    for (int k = 0; k < 10; k++) {
        float s = 0.0f;
        for (int idx = tid; idx < NBLK; idx += BLK) s += part[(size_t)idx * 10 + k];
        #pragma unroll
        for (int off = 16; off > 0; off >>= 1) s += __shfl_down(s, off, 32);
        if ((tid & 31) == 0) sW[tid >> 5] = s;
        __syncthreads();
        if (tid == 0) {
            float t = 0.0f;
            #pragma unroll
            for (int w = 0; w < 8; w++) t += sW[w];
            sAcc[k] = t;
        }
        __syncthreads();
    }
    if (tid == 0) {
        float lx = sAcc[0], ly = sAcc[1], lw = sAcc[2], lh = sAcc[3], lc = sAcc[4];
        float lcl = sAcc[5], lyw = sAcc[6], lpt = sAcc[7], lrl = sAcc[8], np = sAcc[9];
        out[0]  = lx;  out[1] = ly;  out[2] = lw;  out[3] = lh;
        out[4]  = lc;  out[5] = lcl; out[6] = lyw; out[7] = lpt; out[8] = lrl;
        out[9]  = lx + ly + (lw + lh) + lc + 0.5f * lcl;
        out[10] = (4.0f / 3.0f) * (lyw + lpt + lrl);
        out[11] = fmaxf(np, 1.0f);
    }
}

// ---------------- launch ----------------
extern "C" void kernel_launch(void* const* d_in, const int* in_sizes, int n_in,
                              void* d_out, int out_size, void* d_ws, size_t ws_size,
                              hipStream_t stream) {
    const float* inp = (const float*)d_in[0];   // (32, 27, 76, 76) f32
    const float* tgt = (const float*)d_in[1];   // (32, 40, 8) f32
    float* out = (float*)d_out;                 // 12 f32

    // workspace layout (floats): gt[7680] | rec[15360] | part[21760]
    float* ws   = (float*)d_ws;
    float* gt   = ws;
    float* rec  = ws + BS * NT * GTS;                     // +7680
    float* part = ws + BS * NT * GTS + BS * NT * 12;      // +23040

    k_prep<<<(BS * NT + BLK - 1) / BLK, BLK, 0, stream>>>(tgt, gt, rec);
    k_main<<<NBLK, BLK, 0, stream>>>(inp, gt, rec, part);
    k_final<<<1, BLK, 0, stream>>>(part, out);
}